// CRFDependency_74689481277823
// MI455X (gfx1250) — compile-verified
//
#include <hip/hip_runtime.h>
#include <hip/hip_bf16.h>
#include <stdint.h>

#define LL 128          // sequence length L
#define BB 64           // batch B
#define NTHR 128        // 4 wave32 waves per workgroup
#define NEG_INF (-__builtin_inff())

typedef unsigned int u32x4 __attribute__((ext_vector_type(4)));
typedef int          i32x4 __attribute__((ext_vector_type(4)));
typedef int          i32x8 __attribute__((ext_vector_type(8)));
typedef float        v2f   __attribute__((ext_vector_type(2)));
typedef float        v8f   __attribute__((ext_vector_type(8)));

// Dynamic LDS layout (bytes):
//   [0,      65536)  s_i  table (L*L f32)
//   [65536, 131072)  s_c  table (L*L f32)
//   [131072,196608)  sc   = scores[b] tile (L*L f32), filled by TDM
//   [196608,197120)  red  scratch (128 f32)
//   [197120,197124)  s_len
#define SMEM_BYTES 197632

__global__ __launch_bounds__(NTHR)
void crf_inside_kernel(const float* __restrict__ scores,
                       const int*   __restrict__ mask,
                       const int*   __restrict__ target,
                       float*       __restrict__ ws) {
  extern __shared__ char smem[];
  float* s_i  = (float*)(smem);
  float* s_c  = (float*)(smem + 65536);
  float* sc   = (float*)(smem + 131072);
  float* red  = (float*)(smem + 196608);
  int*   s_ln = (int*)  (smem + 197120);

  const int b   = blockIdx.x;
  const int tid = threadIdx.x;

  // ---- CDNA5 Tensor Data Mover: async load scores[b] (128x128 f32, 64KB)
  // into LDS. Descriptor per ISA cdna5 ch.8: group0 = {count, lds_addr,
  // global_addr, type=2}; group1 = {data_size=4B, tensor_dim0/1=128,
  // tile_dim0/1=128, dim0_stride=128}. EXEC is ignored by TENSOR ops, so the
  // guard makes exactly wave 0 issue it once.
  if (tid == 0) {
    *s_ln = 0;
    unsigned long long ga =
        (unsigned long long)(uintptr_t)(scores + (size_t)b * LL * LL);
    u32x4 g0;
    g0[0] = 1u;                                           // count = 1 (valid)
    g0[1] = __builtin_amdgcn_groupstaticsize() + 131072u; // lds_addr of sc[]
    g0[2] = (unsigned)(ga & 0xFFFFFFFFull);               // global_addr[31:0]
    g0[3] = (unsigned)((ga >> 32) & 0x01FFFFFFull)        // global_addr[56:32]
          | (2u << 30);                                   // type = 2 ("image")
    i32x8 g1;
    g1[0] = (int)(2u << 16);      // data_size = 2 -> 4 bytes
    g1[1] = (int)(128u << 16);    // tensor_dim0 = 128 (bits 79:48)
    g1[2] = (int)(128u << 16);    // tensor_dim1 = 128 (bits 111:80)
    g1[3] = (int)(128u << 16);    // tile_dim0   = 128 (bits 127:112)
    g1[4] = 128;                  // tile_dim1 = 128, tile_dim2 = 0 (2D tile)
    g1[5] = 128;                  // tensor_dim0_stride = 128 elements
    g1[6] = 0;                    // stride high bits / dim1_stride = 0
    g1[7] = 0;
    i32x4 gz  = {0, 0, 0, 0};               // D# groups 2/3 unused (2D)
    i32x8 gz8 = {0, 0, 0, 0, 0, 0, 0, 0};   // 6-arg form extra group (zeroed)
    __builtin_amdgcn_tensor_load_to_lds(g0, g1, gz, gz, gz8, 0);
  }

  // Overlap the 64KB TDM stream with table init (128KB of LDS writes).
  for (int idx = tid; idx < LL * LL; idx += NTHR) {
    s_i[idx] = NEG_INF;
    s_c[idx] = NEG_INF;
  }
  for (int d = tid; d < LL; d += NTHR) s_c[d * LL + d] = 0.0f;

  __syncthreads();                       // s_ln zero visible
  if (mask[b * LL + tid] != 0) atomicAdd(s_ln, 1);

  if (tid == 0) __builtin_amdgcn_s_wait_tensorcnt(0);  // TDM done (wave 0)
  __syncthreads();                       // lens + sc visible to all waves
  const int lens = *s_ln;

  // ---- inside recursion: widths are sequential; spans parallel (1/thread).
  // Phase 1 writes s_i (reads s_c widths < w); barrier; phase 2 writes s_c
  // (reads s_c widths < w and s_i widths <= w). Matches reference dataflow.
  for (int w = 1; w < LL; ++w) {
    const int i = tid;
    if (i < LL - w) {
      const float* r0 = s_c + i * LL;             // s_c[i][*]
      const float* r1 = s_c + (i + w) * LL;       // s_c[i+w][*]
      float m = NEG_INF;
      for (int j = 0; j < w; ++j)
        m = fmaxf(m, r0[i + j] + r1[i + j + 1]);
      float il = NEG_INF;
      if (m > NEG_INF) {
        float s = 0.0f;
        for (int j = 0; j < w; ++j)
          s += __expf(r0[i + j] + r1[i + j + 1] - m);
        il = __logf(s) + m;
      }
      // scores_p[lo,hi,b] = scores[b,hi,lo] = sc[hi*L+lo]
      s_i[(i + w) * LL + i] = il + sc[i * LL + (i + w)];
      s_i[i * LL + (i + w)] = il + sc[(i + w) * LL + i];
    }
    __syncthreads();
    if (i < LL - w) {
      // close-left: cl_j = s_c[i+j][i] + s_i[i+w][i+j]
      float m = NEG_INF;
      for (int j = 0; j < w; ++j)
        m = fmaxf(m, s_c[(i + j) * LL + i] + s_i[(i + w) * LL + (i + j)]);
      float clv = NEG_INF;
      if (m > NEG_INF) {
        float s = 0.0f;
        for (int j = 0; j < w; ++j)
          s += __expf(s_c[(i + j) * LL + i] + s_i[(i + w) * LL + (i + j)] - m);
        clv = __logf(s) + m;
      }
      // close-right: cr_j = s_i[i][i+j+1] + s_c[i+j+1][i+w]
      float m2 = NEG_INF;
      for (int j = 0; j < w; ++j)
        m2 = fmaxf(m2, s_i[i * LL + (i + j + 1)] + s_c[(i + j + 1) * LL + (i + w)]);
      float crv = NEG_INF;
      if (m2 > NEG_INF) {
        float s = 0.0f;
        for (int j = 0; j < w; ++j)
          s += __expf(s_i[i * LL + (i + j + 1)] +
                      s_c[(i + j + 1) * LL + (i + w)] - m2);
        crv = __logf(s) + m2;
      }
      if (i == 0 && lens != w) crv = NEG_INF;   // root row gated to lens==w
      s_c[(i + w) * LL + i] = clv;
      s_c[i * LL + (i + w)] = crv;
    }
    __syncthreads();
  }

  // ---- per-batch partials: logZ_b, masked gold sum, lens (fixed-order sum)
  {
    const int tg = target[b * LL + tid];
    red[tid] = (mask[b * LL + tid] != 0) ? sc[tid * LL + tg] : 0.0f;
  }
  __syncthreads();
  if (tid == 0) {
    float gs = 0.0f;
    for (int l = 0; l < LL; ++l) gs += red[l];
    ws[b]           = s_c[0 * LL + lens];   // logZ_b
    ws[BB + b]      = gs;                   // gold_b
    ws[2 * BB + b]  = (float)lens;          // lens_b
  }
}

// Final reduction: three 64-element sums via V_WMMA_F32_16X16X4_F32 with a
// ones A-matrix. D[r][c] = sum_k B[k][c], so sum(all D) = 16 * sum(all B) --
// invariant to the B-fragment lane layout, hence exact regardless of the
// 4x16 striping. One wave32, uniform flow (EXEC all ones for WMMA).
__global__ __launch_bounds__(32)
void crf_finish_kernel(const float* __restrict__ ws, float* __restrict__ out) {
  __shared__ float red[32];
  __shared__ float tot[3];
  const int l = threadIdx.x;
  for (int part = 0; part < 3; ++part) {
    v2f a; a.x = 1.0f; a.y = 1.0f;               // ones A fragment (16x4)
    v2f bfr;                                     // B fragment (4x16) = 64 vals
    bfr.x = ws[part * 64 + 2 * l + 0];
    bfr.y = ws[part * 64 + 2 * l + 1];
    v8f c = {0.f, 0.f, 0.f, 0.f, 0.f, 0.f, 0.f, 0.f};
    v8f d = __builtin_amdgcn_wmma_f32_16x16x4_f32(
        /*neg_a=*/false, a, /*neg_b=*/false, bfr,
        /*c_mod=*/(short)0, c, /*reuse_a=*/false, /*reuse_b=*/false);
    float lanesum = d[0] + d[1] + d[2] + d[3] + d[4] + d[5] + d[6] + d[7];
    red[l] = lanesum;
    __syncthreads();
    if (l == 0) {
      float s = 0.0f;
      for (int k = 0; k < 32; ++k) s += red[k];
      tot[part] = s * (1.0f / 16.0f);
    }
    __syncthreads();
  }
  if (l == 0) out[0] = (tot[0] - tot[1]) / tot[2];
}

extern "C" void kernel_launch(void* const* d_in, const int* in_sizes, int n_in,
                              void* d_out, int out_size, void* d_ws,
                              size_t ws_size, hipStream_t stream) {
  (void)in_sizes; (void)n_in; (void)out_size; (void)ws_size;
  const float* scores = (const float*)d_in[0];
  const int*   mask   = (const int*)d_in[1];
  const int*   target = (const int*)d_in[2];
  float*       out    = (float*)d_out;
  float*       ws     = (float*)d_ws;

  // 192KB+ dynamic LDS per workgroup (CDNA5 WGP has 320KB) -- opt in.
  (void)hipFuncSetAttribute((const void*)crf_inside_kernel,
                            hipFuncAttributeMaxDynamicSharedMemorySize,
                            (int)SMEM_BYTES);

  crf_inside_kernel<<<dim3(BB), dim3(NTHR), SMEM_BYTES, stream>>>(
      scores, mask, target, ws);
  crf_finish_kernel<<<dim3(1), dim3(32), 0, stream>>>(ws, out);
}